// ImprovedRiemannianVAE_23029614641795
// MI455X (gfx1250) — compile-verified
//
#include <hip/hip_runtime.h>
#include <math.h>

// ---------------------------------------------------------------------------
// ImprovedRiemannianVAE forward for MI455X (gfx1250, wave32, WMMA f32 path)
// ---------------------------------------------------------------------------

typedef float v2f __attribute__((ext_vector_type(2)));
typedef float v8f __attribute__((ext_vector_type(8)));

#define N64     64
#define LDA     66            // even pad: 8B-aligned rows (ds_load_b64), 2-way
                              // bank conflict worst case on column phase
#define MSZ     (N64 * LDA)   // 4224 floats per 64x64 LDS matrix
#define BATCH   4096
#define SPD_DIM 2080
#define EPSV    1e-6f
#define BN_EPSV 1e-5f

// ---------------------------------------------------------------------------
// 16x16x4 f32 WMMA fragment helpers (wave32; EXEC all ones in these regions)
// A frag: lanes 0-15 hold M=0..15 {K=k0,k0+1}; lanes 16-31 hold {K=k0+2,k0+3}
// B frag: lanes 0-15 hold N=0..15 {K=k0,k0+1}; lanes 16-31 hold {K=k0+2,k0+3}
// C/D:    VGPR r -> M = r (lanes 0-15) / M = r+8 (lanes 16-31), N = lane&15
// ---------------------------------------------------------------------------

// C(64x64) = A(64x64) @ B(64x64)  or  A @ B^T, all in LDS with leading dim LDA.
// 8 waves per workgroup, each wave does 2 of the 16 output tiles.
// All 16 A/B fragments are preloaded, then 16 WMMAs issue back-to-back.
// Caller must __syncthreads() before (inputs ready) and after (C complete).
__device__ __forceinline__ void mm64(const float* __restrict__ A,
                                     const float* __restrict__ B,
                                     float* __restrict__ C,
                                     int tid, bool bTransposed) {
  const int wave = tid >> 5;
  const int lane = tid & 31;
  const int half = lane >> 4;
  const int l    = lane & 15;
  for (int t = wave; t < 16; t += 8) {
    const int tm = t >> 2, tn = t & 3;
    v2f af[16], bf[16];
    const float* arow = A + (tm * 16 + l) * LDA + 2 * half;  // 8B aligned
#pragma unroll
    for (int kk = 0; kk < 16; ++kk)
      af[kk] = *(const v2f*)(arow + 4 * kk);
    if (bTransposed) {                  // Bop[k][n] = B[n][k]
      const float* brow = B + (tn * 16 + l) * LDA + 2 * half;
#pragma unroll
      for (int kk = 0; kk < 16; ++kk)
        bf[kk] = *(const v2f*)(brow + 4 * kk);
    } else {                            // Bop[k][n] = B[k][n]
#pragma unroll
      for (int kk = 0; kk < 16; ++kk) {
        const int k = 4 * kk + 2 * half;
        v2f b;
        b.x = B[(k)     * LDA + tn * 16 + l];
        b.y = B[(k + 1) * LDA + tn * 16 + l];
        bf[kk] = b;
      }
    }
    v8f acc = {};
#pragma unroll
    for (int kk = 0; kk < 16; ++kk)
      acc = __builtin_amdgcn_wmma_f32_16x16x4_f32(false, af[kk], false, bf[kk],
                                                  (short)0, acc, false, false);
#pragma unroll
    for (int r = 0; r < 8; ++r)
      C[(tm * 16 + r + 8 * half) * LDA + tn * 16 + l] = acc[r];
  }
}

// ---------------------------------------------------------------------------
// Parallel cyclic Jacobi eigensolver on a 64x64 symmetric LDS matrix.
// Round-robin tournament: 32 disjoint (p,q) rotations per round, 63 rounds
// per sweep. Row phase (J^T A, vectorized v2f) then column phase (A J) with
// barriers; eigenvector accumulation V <- V J shares the column phase.
// On exit: diag(A) = eigenvalues, V columns = eigenvectors (if V != nullptr).
// 256 threads per workgroup; 8 threads per rotation pair.
// ---------------------------------------------------------------------------
__device__ void jacobi64(float* __restrict__ A, float* __restrict__ V,
                         float* __restrict__ csv, int* __restrict__ prq,
                         int tid, int sweeps) {
  if (V != nullptr) {
    for (int t = tid; t < N64 * N64; t += 256) {
      int i = t >> 6, j = t & 63;
      V[i * LDA + j] = (i == j) ? 1.0f : 0.0f;
    }
  }
  __syncthreads();

  for (int sw = 0; sw < sweeps; ++sw) {
    for (int r = 0; r < 63; ++r) {
      if (tid < 32) {
        int i = tid, j = 63 - tid;
        int p = (i == 0) ? 0 : 1 + ((i - 1 + r) % 63);
        int q = 1 + ((j - 1 + r) % 63);
        if (p > q) { int tmp = p; p = q; q = tmp; }
        float app = A[p * LDA + p];
        float aqq = A[q * LDA + q];
        float apq = A[p * LDA + q];
        float c = 1.0f, s = 0.0f;
        if (fabsf(apq) > 1e-12f) {
          float theta = (aqq - app) / (2.0f * apq);
          float tt = copysignf(1.0f, theta) /
                     (fabsf(theta) + sqrtf(theta * theta + 1.0f));
          c = 1.0f / sqrtf(tt * tt + 1.0f);
          s = tt * c;
        }
        prq[2 * tid] = p; prq[2 * tid + 1] = q;
        csv[2 * tid] = c; csv[2 * tid + 1] = s;
      }
      __syncthreads();

      const int pk   = tid >> 3;           // pair 0..31
      const int p    = prq[2 * pk];
      const int q    = prq[2 * pk + 1];
      const float c  = csv[2 * pk];
      const float s  = csv[2 * pk + 1];
      const int base = (tid & 7) * 8;      // 8 contiguous elements per thread

      // Row phase: A <- J^T A (rows disjoint across pairs), v2f b64 accesses
      {
        float* Ap = A + p * LDA + base;    // 8B aligned (LDA even, base % 8)
        float* Aq = A + q * LDA + base;
#pragma unroll
        for (int m = 0; m < 8; m += 2) {
          v2f ap = *(v2f*)(Ap + m);
          v2f aq = *(v2f*)(Aq + m);
          *(v2f*)(Ap + m) = c * ap - s * aq;
          *(v2f*)(Aq + m) = s * ap + c * aq;
        }
      }
      __syncthreads();

      // Column phase: A <- A J ; V <- V J (columns disjoint across pairs)
#pragma unroll
      for (int m = 0; m < 8; ++m) {
        int i = base + m;
        float ap = A[i * LDA + p], aq = A[i * LDA + q];
        A[i * LDA + p] = c * ap - s * aq;
        A[i * LDA + q] = s * ap + c * aq;
        if (V != nullptr) {
          float vp = V[i * LDA + p], vq = V[i * LDA + q];
          V[i * LDA + p] = c * vp - s * vq;
          V[i * LDA + q] = s * vp + c * vq;
        }
      }
      __syncthreads();
    }
  }
}

// shift = max(EPS - min(diag(A)), 0), broadcast via LDS
__device__ float spd_shift(const float* __restrict__ A, float* __restrict__ wred,
                           int tid) {
  if (tid < 64) wred[tid] = A[tid * LDA + tid];
  __syncthreads();
  if (tid == 0) {
    float mn = wred[0];
    for (int i = 1; i < 64; ++i) mn = fminf(mn, wred[i]);
    wred[0] = fmaxf(EPSV - mn, 0.0f);
  }
  __syncthreads();
  float sh = wred[0];
  __syncthreads();
  return sh;
}

// triu_indices(64) linear index -> (i, j)
__device__ __forceinline__ void triu_ij(int t, int& i, int& j) {
  int off = 0; i = 0;
  while (t >= off + (N64 - i)) { off += N64 - i; ++i; }
  j = i + (t - off);
}

// LDS partition for the per-matrix kernels
#define SMEM_FLOATS (4 * MSZ + 128)
#define SMEM_BYTES  (SMEM_FLOATS * sizeof(float) + 64 * sizeof(int))

// ---------------------------------------------------------------------------
// Stage A (1 workgroup): eigh(ref + eps I) -> rm = ref^-1/2, rp = ref^+1/2
// ---------------------------------------------------------------------------
__global__ void __launch_bounds__(256)
ref_prep_kernel(const float* __restrict__ ref, float* __restrict__ rmrp) {
  extern __shared__ float sm[];
  float* M0 = sm;                 // ref_e -> eigenvalues on diag
  float* M1 = sm + MSZ;           // V
  float* M2 = sm + 2 * MSZ;       // V * f(w)
  float* M3 = sm + 3 * MSZ;       // result
  float* wred = sm + 4 * MSZ;     // 64
  float* csv  = wred + 64;        // 64
  int*   prq  = (int*)(sm + SMEM_FLOATS);
  const int tid = threadIdx.x;

  for (int t = tid; t < N64 * N64; t += 256) {
    int i = t >> 6, j = t & 63;
    M0[i * LDA + j] = ref[t] + ((i == j) ? EPSV : 0.0f);
  }
  __syncthreads();

  jacobi64(M0, M1, csv, prq, tid, 12);

  if (tid < 64) wred[tid] = M0[tid * LDA + tid];
  __syncthreads();

  // rm = V diag(w^-1/2) V^T
  for (int t = tid; t < N64 * N64; t += 256) {
    int i = t >> 6, j = t & 63;
    M2[i * LDA + j] = M1[i * LDA + j] * rsqrtf(fmaxf(wred[j], 1e-30f));
  }
  __syncthreads();
  mm64(M2, M1, M3, tid, /*bT=*/true);
  __syncthreads();
  for (int t = tid; t < N64 * N64; t += 256)
    rmrp[t] = M3[(t >> 6) * LDA + (t & 63)];
  __syncthreads();

  // rp = V diag(w^+1/2) V^T
  for (int t = tid; t < N64 * N64; t += 256) {
    int i = t >> 6, j = t & 63;
    M2[i * LDA + j] = M1[i * LDA + j] * sqrtf(fmaxf(wred[j], 0.0f));
  }
  __syncthreads();
  mm64(M2, M1, M3, tid, /*bT=*/true);
  __syncthreads();
  for (int t = tid; t < N64 * N64; t += 256)
    rmrp[N64 * N64 + t] = M3[(t >> 6) * LDA + (t & 63)];
}

// ---------------------------------------------------------------------------
// Stage B (one workgroup per batch element): ensure_spd, whiten, eigh, log,
// tangent map, upper-tri vectorize.
// ---------------------------------------------------------------------------
__global__ void __launch_bounds__(256)
spd_encode_kernel(const float* __restrict__ x, const float* __restrict__ rmrp,
                  float* __restrict__ vec) {
  extern __shared__ float sm[];
  float* M0 = sm;
  float* M1 = sm + MSZ;
  float* M2 = sm + 2 * MSZ;
  float* M3 = sm + 3 * MSZ;
  float* wred = sm + 4 * MSZ;
  float* csv  = wred + 64;
  int*   prq  = (int*)(sm + SMEM_FLOATS);
  const int tid = threadIdx.x;
  const size_t b = blockIdx.x;
  const float* xb = x + b * (size_t)(N64 * N64);

  // sym_lower(x) into M0 (kept) and M1 (consumed by eigvalsh)
  for (int t = tid; t < N64 * N64; t += 256) {
    int i = t >> 6, j = t & 63;
    float v = (i >= j) ? xb[i * 64 + j] : xb[j * 64 + i];
    M0[i * LDA + j] = v;
    M1[i * LDA + j] = v;
  }
  __syncthreads();

  jacobi64(M1, nullptr, csv, prq, tid, 8);        // eigenvalues only
  float shift = spd_shift(M1, wred, tid);
  if (tid < 64) M0[tid * LDA + tid] += shift;     // x_spd

  // load rm
  for (int t = tid; t < N64 * N64; t += 256)
    M2[(t >> 6) * LDA + (t & 63)] = rmrp[t];
  __syncthreads();

  mm64(M2, M0, M3, tid, false);                   // M3 = rm @ x_spd
  __syncthreads();
  mm64(M3, M2, M1, tid, false);                   // M1 = s
  __syncthreads();

  jacobi64(M1, M3, csv, prq, tid, 10);            // diag(M1)=w, M3=V

  // M0 = V * log(max(w,1e-12)) (column scale)
  for (int t = tid; t < N64 * N64; t += 256) {
    int i = t >> 6, j = t & 63;
    M0[i * LDA + j] = M3[i * LDA + j] * logf(fmaxf(M1[j * LDA + j], 1e-12f));
  }
  __syncthreads();
  mm64(M0, M3, M1, tid, /*bT=*/true);             // M1 = log_s
  __syncthreads();
  mm64(M2, M1, M0, tid, false);                   // M0 = rm @ log_s
  __syncthreads();

  // load rp over rm (rm no longer needed)
  for (int t = tid; t < N64 * N64; t += 256)
    M2[(t >> 6) * LDA + (t & 63)] = rmrp[N64 * N64 + t];
  __syncthreads();
  mm64(M0, M2, M1, tid, false);                   // M1 = tang
  __syncthreads();

  // upper-tri vectorize (np.triu_indices order)
  for (int t = tid; t < SPD_DIM; t += 256) {
    int i, j; triu_ij(t, i, j);
    vec[b * (size_t)SPD_DIM + t] = M1[i * LDA + j];
  }
}

// ---------------------------------------------------------------------------
// Batched GEMM: C(M x N) = epilogue(A(M x K) @ W(N x K)^T + bias)
// mode 0: linear. mode 1: leaky(0.2) then BN: h * rsqrt(1+eps) * g + b.
// One wave per 16x16 output tile; M=4096, N,K multiples of 16/4.
// Fragments loaded as v2f (global_load_b64), WMMA chained on accumulator.
// ---------------------------------------------------------------------------
__global__ void __launch_bounds__(32)
gemm_wmma_f32(const float* __restrict__ A, const float* __restrict__ W,
              const float* __restrict__ bias, const float* __restrict__ g,
              const float* __restrict__ bb, float* __restrict__ C,
              int K, int N, int mode) {
  const int tn = blockIdx.x, tm = blockIdx.y;
  const int lane = threadIdx.x;
  const int half = lane >> 4;
  const int l    = lane & 15;
  const float* arow = A + (size_t)(tm * 16 + l) * K + 2 * half;  // 8B aligned
  const float* wrow = W + (size_t)(tn * 16 + l) * K + 2 * half;

  v8f acc = {};
  for (int k0 = 0; k0 < K; k0 += 4) {
    v2f a = *(const v2f*)(arow + k0);
    v2f b = *(const v2f*)(wrow + k0);
    __builtin_prefetch(arow + k0 + 64, 0, 1);
    acc = __builtin_amdgcn_wmma_f32_16x16x4_f32(false, a, false, b,
                                                (short)0, acc, false, false);
  }

  const int col = tn * 16 + l;
  const float invs = rsqrtf(1.0f + BN_EPSV);
  const float bi = bias[col];
#pragma unroll
  for (int r = 0; r < 8; ++r) {
    const int row = tm * 16 + r + 8 * half;
    float v = acc[r] + bi;
    if (mode == 1) {
      v = (v >= 0.0f) ? v : 0.2f * v;
      v = v * invs * g[col] + bb[col];
    }
    C[(size_t)row * N + col] = v;
  }
}

// ---------------------------------------------------------------------------
// Stage D (one workgroup per batch element): rebuild symmetric matrix, eigh,
// matrix exp, ref @ exp_a, ensure_spd, write recon.
// ---------------------------------------------------------------------------
__global__ void __launch_bounds__(256)
spd_decode_kernel(const float* __restrict__ vdec, const float* __restrict__ ref,
                  float* __restrict__ recon) {
  extern __shared__ float sm[];
  float* M0 = sm;
  float* M1 = sm + MSZ;
  float* M2 = sm + 2 * MSZ;
  float* M3 = sm + 3 * MSZ;
  float* wred = sm + 4 * MSZ;
  float* csv  = wred + 64;
  int*   prq  = (int*)(sm + SMEM_FLOATS);
  const int tid = threadIdx.x;
  const size_t b = blockIdx.x;
  const float* vb = vdec + b * (size_t)SPD_DIM;

  // m + eps*I into M0 (symmetric fill from upper-tri vector)
  for (int t = tid; t < SPD_DIM; t += 256) {
    int i, j; triu_ij(t, i, j);
    float v = vb[t];
    M0[i * LDA + j] = v;
    M0[j * LDA + i] = v;
  }
  __syncthreads();
  if (tid < 64) M0[tid * LDA + tid] += EPSV;
  __syncthreads();

  jacobi64(M0, M1, csv, prq, tid, 10);            // diag(M0)=w, M1=V

  // M2 = V * exp(w) (column scale)
  for (int t = tid; t < N64 * N64; t += 256) {
    int i = t >> 6, j = t & 63;
    M2[i * LDA + j] = M1[i * LDA + j] * expf(M0[j * LDA + j]);
  }
  __syncthreads();
  mm64(M2, M1, M3, tid, /*bT=*/true);             // M3 = exp_a
  __syncthreads();

  // load ref
  for (int t = tid; t < N64 * N64; t += 256)
    M0[(t >> 6) * LDA + (t & 63)] = ref[t];
  __syncthreads();
  mm64(M0, M3, M2, tid, false);                   // M2 = ref @ exp_a
  __syncthreads();

  // ensure_spd: eigvalsh(sym_lower(M2)) -> shift
  for (int t = tid; t < N64 * N64; t += 256) {
    int i = t >> 6, j = t & 63;
    M1[i * LDA + j] = (i >= j) ? M2[i * LDA + j] : M2[j * LDA + i];
  }
  __syncthreads();
  jacobi64(M1, nullptr, csv, prq, tid, 8);
  float shift = spd_shift(M1, wred, tid);

  float* rb = recon + b * (size_t)(N64 * N64);
  for (int t = tid; t < N64 * N64; t += 256) {
    int i = t >> 6, j = t & 63;
    rb[t] = M2[i * LDA + j] + ((i == j) ? shift : 0.0f);
  }
}

// ---------------------------------------------------------------------------
// Host launcher
// ---------------------------------------------------------------------------
extern "C" void kernel_launch(void* const* d_in, const int* in_sizes, int n_in,
                              void* d_out, int out_size, void* d_ws, size_t ws_size,
                              hipStream_t stream) {
  (void)in_sizes; (void)n_in; (void)out_size; (void)ws_size;

  const float* x      = (const float*)d_in[0];
  const float* ref    = (const float*)d_in[1];
  const float* enc_w1 = (const float*)d_in[2];
  const float* enc_b1 = (const float*)d_in[3];
  const float* bn1_g  = (const float*)d_in[4];
  const float* bn1_b  = (const float*)d_in[5];
  const float* enc_w2 = (const float*)d_in[6];
  const float* enc_b2 = (const float*)d_in[7];
  const float* bn2_g  = (const float*)d_in[8];
  const float* bn2_b  = (const float*)d_in[9];
  const float* mu_w   = (const float*)d_in[10];
  const float* mu_b   = (const float*)d_in[11];
  const float* lv_w   = (const float*)d_in[12];
  const float* lv_b   = (const float*)d_in[13];
  const float* dec_w1 = (const float*)d_in[14];
  const float* dec_b1 = (const float*)d_in[15];
  const float* dbn1_g = (const float*)d_in[16];
  const float* dbn1_b = (const float*)d_in[17];
  const float* dec_w2 = (const float*)d_in[18];
  const float* dec_b2 = (const float*)d_in[19];
  const float* dbn2_g = (const float*)d_in[20];
  const float* dbn2_b = (const float*)d_in[21];
  const float* dec_w3 = (const float*)d_in[22];
  const float* dec_b3 = (const float*)d_in[23];

  float* out   = (float*)d_out;
  float* recon = out;                                   // 4096*64*64
  float* mu    = out + (size_t)BATCH * N64 * N64;       // 4096*32
  float* lv    = mu + (size_t)BATCH * 32;               // 4096*32

  float* ws   = (float*)d_ws;
  float* rmrp = ws;                                     // 2 * 64*64
  float* vec  = rmrp + 2 * (size_t)(N64 * N64);         // 4096*2080
  float* h1   = vec + (size_t)BATCH * SPD_DIM;          // 4096*128
  float* h2   = h1 + (size_t)BATCH * 128;               // 4096*64
  float* dd1  = h2 + (size_t)BATCH * 64;                // 4096*64
  float* dd2  = dd1 + (size_t)BATCH * 64;               // 4096*128
  float* vdec = dd2 + (size_t)BATCH * 128;              // 4096*2080

  const size_t smem = SMEM_BYTES;
  const dim3 wg256(256);

  ref_prep_kernel<<<1, wg256, smem, stream>>>(ref, rmrp);
  spd_encode_kernel<<<BATCH, wg256, smem, stream>>>(x, rmrp, vec);

  gemm_wmma_f32<<<dim3(128 / 16, BATCH / 16), 32, 0, stream>>>(
      vec, enc_w1, enc_b1, bn1_g, bn1_b, h1, SPD_DIM, 128, 1);
  gemm_wmma_f32<<<dim3(64 / 16, BATCH / 16), 32, 0, stream>>>(
      h1, enc_w2, enc_b2, bn2_g, bn2_b, h2, 128, 64, 1);
  gemm_wmma_f32<<<dim3(32 / 16, BATCH / 16), 32, 0, stream>>>(
      h2, mu_w, mu_b, nullptr, nullptr, mu, 64, 32, 0);
  gemm_wmma_f32<<<dim3(32 / 16, BATCH / 16), 32, 0, stream>>>(
      h2, lv_w, lv_b, nullptr, nullptr, lv, 64, 32, 0);
  gemm_wmma_f32<<<dim3(64 / 16, BATCH / 16), 32, 0, stream>>>(
      mu, dec_w1, dec_b1, dbn1_g, dbn1_b, dd1, 32, 64, 1);
  gemm_wmma_f32<<<dim3(128 / 16, BATCH / 16), 32, 0, stream>>>(
      dd1, dec_w2, dec_b2, dbn2_g, dbn2_b, dd2, 64, 128, 1);
  gemm_wmma_f32<<<dim3(SPD_DIM / 16, BATCH / 16), 32, 0, stream>>>(
      dd2, dec_w3, dec_b3, nullptr, nullptr, vdec, 128, SPD_DIM, 0);

  spd_decode_kernel<<<BATCH, wg256, smem, stream>>>(vdec, ref, recon);
}